// CTCLayer_3427383902291
// MI455X (gfx1250) — compile-verified
//
#include <hip/hip_runtime.h>

// Problem constants (from the reference)
#define BB 256
#define TT 256
#define CC 512
#define LL 64
#define SS (2 * LL + 1)   // 129 extended states
#define BLANK (CC - 1)
#define NEGF (-1e30f)
#define EPSF (1e-7f)
#define NBUF 4            // async row ring buffer depth (3 rows in flight)

#if __has_builtin(__builtin_amdgcn_global_load_async_to_lds_b128)
#define HAVE_ASYNC_LDS 1
#endif

typedef int v4i __attribute__((ext_vector_type(4)));
typedef __attribute__((address_space(1))) v4i* gv4p;  // global pointer to 16B chunk
typedef __attribute__((address_space(3))) v4i* lv4p;  // LDS pointer to 16B chunk

// One 16-byte chunk: global -> LDS. Async (ASYNCcnt) on gfx1250.
__device__ __forceinline__ void row_cp16(const float* g, float* l) {
#ifdef HAVE_ASYNC_LDS
  __builtin_amdgcn_global_load_async_to_lds_b128((gv4p)g, (lv4p)l, 0, 0);
#else
  *(float4*)l = *(const float4*)g;
#endif
}

// Wait until this wave's outstanding async-LDS loads <= n (literal).
#ifdef HAVE_ASYNC_LDS
#if __has_builtin(__builtin_amdgcn_s_wait_asynccnt)
#define WAIT_ASYNC_LE(n) __builtin_amdgcn_s_wait_asynccnt(n)
#else
#define WAIT_ASYNC_LE(n) asm volatile("s_wait_asynccnt " #n ::: "memory")
#endif
#else
#define WAIT_ASYNC_LE(n) do {} while (0)
#endif

__global__ __launch_bounds__(256) void ctc_batch_cost_kernel(
    const int* __restrict__ y_true,   // [B, L+1] int32: labels + label_len
    const float* __restrict__ y_pred, // [B, T, C] softmax probabilities
    float* __restrict__ out)          // [B, 1] loss
{
  __shared__ __align__(16) float row[NBUF][CC];  // 8 KB prob-row ring buffer
  __shared__ float alpha[2][SS + 3];             // [0..1] = -inf guards, states at +2

  const int b   = blockIdx.x;
  const int tid = threadIdx.x;
  const float* __restrict__ base = y_pred + (size_t)b * TT * CC;
  const int*   __restrict__ yt   = y_true + b * (LL + 1);

  // Issue rows 0..NBUF-2 into the ring (waves 0..3, one b128 per lane per row).
  if (tid < CC / 4) {
#pragma unroll
    for (int r = 0; r < NBUF - 1; ++r)
      row_cp16(base + (size_t)r * CC + tid * 4, &row[r][tid * 4]);
  }

  // Extended label sequence + skip mask, kept in registers (each thread only
  // ever needs its own state's values): odd s = label, even s = blank.
  int  e  = BLANK;
  bool sk = false;
  if (tid < SS && (tid & 1)) {
    const int j = (tid - 1) >> 1;
    e = yt[j];
    if (tid >= 3) sk = (yt[j] != yt[j - 1]);
  }
  if (tid < 2) { alpha[0][tid] = NEGF; alpha[1][tid] = NEGF; }

  WAIT_ASYNC_LE(2);       // row 0 landed (rows 1,2 may still be in flight)
  __syncthreads();

  // t = 0 init into alpha buffer 0.
  if (tid < SS) {
    float a = NEGF;
    if (tid < 2) a = __logf(row[0][e] + EPSF);
    alpha[0][2 + tid] = a;
  }
  // Top off the pipeline: row NBUF-1.
  if (tid < CC / 4)
    row_cp16(base + (size_t)(NBUF - 1) * CC + tid * 4, &row[NBUF - 1][tid * 4]);

  for (int t = 1; t < TT; ++t) {
    const int cur = t & 1, prv = cur ^ 1;
    WAIT_ASYNC_LE(2);     // oldest outstanding row (row t) has landed
    __syncthreads();      // all waves' chunks visible; everyone done with row t-1 & alpha step t-1
    // Refill: row t+NBUF-1 overwrites the slot that held row t-1 (free as of this barrier).
    if (t + NBUF - 1 < TT && tid < CC / 4)
      row_cp16(base + (size_t)(t + NBUF - 1) * CC + tid * 4,
               &row[(t + NBUF - 1) % NBUF][tid * 4]);
    if (tid < SS) {
      const float* ap = alpha[prv];
      const float a0 = ap[2 + tid];
      const float a1 = ap[1 + tid];
      const float a2 = sk ? ap[tid] : NEGF;
      const float m  = fmaxf(a0, fmaxf(a1, a2));
      const float s  = __expf(a0 - m) + __expf(a1 - m) + __expf(a2 - m);
      const float lp = __logf(row[t % NBUF][e] + EPSF);
      alpha[cur][2 + tid] = m + __logf(s) + lp;
    }
  }
  __syncthreads();

  if (tid == 0) {
    const float* aF = alpha[(TT - 1) & 1];
    const int last  = 2 * yt[LL];              // label_len in [L/2, L] -> last <= 128
    const float x = aF[2 + last];
    const float y = aF[1 + last];
    const float m = fmaxf(x, y);
    out[b] = -(m + __logf(__expf(x - m) + __expf(y - m)));
  }
}

extern "C" void kernel_launch(void* const* d_in, const int* in_sizes, int n_in,
                              void* d_out, int out_size, void* d_ws, size_t ws_size,
                              hipStream_t stream) {
  (void)in_sizes; (void)n_in; (void)out_size; (void)d_ws; (void)ws_size;
  const int*   y_true = (const int*)d_in[0];
  const float* y_pred = (const float*)d_in[1];
  float*       out    = (float*)d_out;
  ctc_batch_cost_kernel<<<dim3(BB), dim3(256), 0, stream>>>(y_true, y_pred, out);
}